// GraphConvolution_54434415510061
// MI455X (gfx1250) — compile-verified
//
#include <hip/hip_runtime.h>

typedef __attribute__((ext_vector_type(2))) float v2f;
typedef __attribute__((ext_vector_type(8))) float v8f;

#define D_DIM 64   // D_IN == D_OUT == 64

// ---------------------------------------------------------------------------
// GEMM: H[M x 64] = X[M x 64] @ Wr[64 x 64], fp32, via V_WMMA_F32_16X16X4_F32.
// One wave32 computes a 16-row x 64-col strip: 4 accumulators (v8f each),
// K=64 consumed in 16 steps of K=4.
// W is staged in LDS PRE-PACKED in B-fragment order:
//   lds_w2[k/2][n] = { W[k][n], W[k+1][n] }  (float2)
// so each B-fragment is one aligned ds_load_b64 straight into an even VGPR
// pair (no v_mov repacking before v_wmma).
// Block = 128 threads = 4 waves = 64 rows. M (=B*N=200000) % 64 == 0.
// ---------------------------------------------------------------------------
__global__ void __launch_bounds__(128)
rgcn_gemm_wmma(const float* __restrict__ X, const float* __restrict__ Wr,
               float* __restrict__ H)
{
    __shared__ v2f lds_w2[(D_DIM / 2) * D_DIM];   // 32 x 64 float2 = 16 KB

    // Cooperative packed fill: slot i = (k2, n); two coalesced b32 streams.
    for (int i = threadIdx.x; i < (D_DIM / 2) * D_DIM; i += blockDim.x) {
        const int k2 = i >> 6;          // k/2
        const int n  = i & 63;
        v2f p;
        p.x = Wr[(2 * k2 + 0) * D_DIM + n];
        p.y = Wr[(2 * k2 + 1) * D_DIM + n];
        lds_w2[i] = p;
    }
    __syncthreads();

    const int lane  = threadIdx.x & 31;
    const int wave  = threadIdx.x >> 5;
    const int row0  = blockIdx.x * 64 + wave * 16;
    const int mIdx  = lane & 15;           // M (A) / N (B) index within half-wave
    const int koff  = (lane >> 4) << 1;    // A/B frag: lanes 16-31 hold K=2,3
    const int khalf = lane >> 4;           // koff/2 for packed-LDS row index
    const int rowHi = (lane >> 4) << 3;    // C/D frag: lanes 16-31 hold M=8..15

    v8f acc0 = {}, acc1 = {}, acc2 = {}, acc3 = {};

    const float* arow = X + (size_t)(row0 + mIdx) * D_DIM + koff;

    #pragma unroll
    for (int k = 0; k < D_DIM; k += 4) {
        // A fragment: 16x4 f32 -> 2 VGPRs (one global_load_b64 per lane)
        v2f a = *(const v2f*)(arow + k);

        // B fragments: one ds_load_b64 each, already in WMMA register order.
        const v2f* brow = &lds_w2[((k >> 1) + khalf) * D_DIM + mIdx];
        v2f b0 = brow[0];
        v2f b1 = brow[16];
        v2f b2 = brow[32];
        v2f b3 = brow[48];

        // (neg_a, A, neg_b, B, c_mod, C, reuse_a, reuse_b)
        acc0 = __builtin_amdgcn_wmma_f32_16x16x4_f32(false, a, false, b0, (short)0, acc0, false, false);
        acc1 = __builtin_amdgcn_wmma_f32_16x16x4_f32(false, a, false, b1, (short)0, acc1, false, false);
        acc2 = __builtin_amdgcn_wmma_f32_16x16x4_f32(false, a, false, b2, (short)0, acc2, false, false);
        acc3 = __builtin_amdgcn_wmma_f32_16x16x4_f32(false, a, false, b3, (short)0, acc3, false, false);
    }

    // Store per C/D layout: VGPR i, lanes 0-15 -> row i, lanes 16-31 -> row 8+i.
    float* hbase = H + (size_t)(row0 + rowHi) * D_DIM + mIdx;
    #pragma unroll
    for (int i = 0; i < 8; ++i) {
        hbase[(size_t)i * D_DIM +  0] = acc0[i];
        hbase[(size_t)i * D_DIM + 16] = acc1[i];
        hbase[(size_t)i * D_DIM + 32] = acc2[i];
        hbase[(size_t)i * D_DIM + 48] = acc3[i];
    }
}

// ---------------------------------------------------------------------------
// Scatter: one block per edge; 256 threads = (b in 0..3) x (d in 0..63).
// out[b, dst, d] += H[b*N + src, d]  via fp32 global atomics (L2-resident:
// out (51 MB) + H (51 MB) both fit the 192 MB L2, HBM sees only writeback).
// ---------------------------------------------------------------------------
__global__ void __launch_bounds__(256)
rgcn_scatter(const float* __restrict__ H, const int* __restrict__ edges,
             float* __restrict__ out, int N)
{
    const int e   = blockIdx.x;
    const int src = edges[2 * e + 0];
    const int dst = edges[2 * e + 1];
    const int b   = threadIdx.x >> 6;
    const int d   = threadIdx.x & 63;

    const float v = H[((size_t)b * N + src) * D_DIM + d];
    unsafeAtomicAdd(&out[((size_t)b * N + dst) * D_DIM + d], v);
}

// ---------------------------------------------------------------------------
// Fallback (only if workspace is too small): recompute x[b,src] @ W[r] per
// edge directly (4x the FLOPs, zero scratch).
// ---------------------------------------------------------------------------
__global__ void __launch_bounds__(256)
rgcn_fused_fallback(const float* __restrict__ X, const float* __restrict__ Wr,
                    const int* __restrict__ edges, float* __restrict__ out, int N)
{
    const int e   = blockIdx.x;
    const int src = edges[2 * e + 0];
    const int dst = edges[2 * e + 1];
    const int b   = threadIdx.x >> 6;
    const int d   = threadIdx.x & 63;

    const float* xrow = X + ((size_t)b * N + src) * D_DIM;
    float s = 0.f;
    #pragma unroll
    for (int k = 0; k < D_DIM; ++k)
        s = fmaf(xrow[k], Wr[k * D_DIM + d], s);
    unsafeAtomicAdd(&out[((size_t)b * N + dst) * D_DIM + d], s);
}

extern "C" void kernel_launch(void* const* d_in, const int* in_sizes, int n_in,
                              void* d_out, int out_size, void* d_ws, size_t ws_size,
                              hipStream_t stream)
{
    const float* x     = (const float*)d_in[0];  // (B, N, 64) f32
    const int*   edges = (const int*)d_in[1];    // (R, E, 2) i32
    const float* W     = (const float*)d_in[2];  // (R, 64, 64) f32

    float* out = (float*)d_out;

    const int B = 4;
    const int N = 50000;
    const int R = in_sizes[2] / (D_DIM * D_DIM);          // 8
    const int E = in_sizes[1] / (R * 2);                  // 200000
    const int M = in_sizes[0] / D_DIM;                    // B*N = 200000 rows

    (void)B;

    hipMemsetAsync(d_out, 0, (size_t)out_size * sizeof(float), stream);

    const size_t hBytes = (size_t)M * D_DIM * sizeof(float);  // 51.2 MB
    if (ws_size >= hBytes) {
        float* h = (float*)d_ws;
        for (int r = 0; r < R; ++r) {
            rgcn_gemm_wmma<<<M / 64, 128, 0, stream>>>(
                x, W + (size_t)r * D_DIM * D_DIM, h);
            rgcn_scatter<<<E, 256, 0, stream>>>(
                h, edges + (size_t)r * E * 2, out, N);
        }
    } else {
        for (int r = 0; r < R; ++r) {
            rgcn_fused_fallback<<<E, 256, 0, stream>>>(
                x, W + (size_t)r * D_DIM * D_DIM,
                edges + (size_t)r * E * 2, out, N);
        }
    }
}